// QuantumConvolution_65481071396089
// MI455X (gfx1250) — compile-verified
//
#include <hip/hip_runtime.h>

// CDNA5 / gfx1250: wave32. WMMA fp32 16x16x4 used for the probs -> <Z> projection
// (exact fp32 GEMM, K=4 matches the factorized pair-prob dimension).

typedef __attribute__((ext_vector_type(2))) float v2f;
typedef __attribute__((ext_vector_type(8))) float v8f;

namespace {
constexpr int kWave = 32;
constexpr int kWaves = 8;
constexpr int kBlock = kWave * kWaves;          // 256 threads = 8 waves
constexpr float kHalfPi = 1.57079632679489661923f;
}

// sin/cos on the restricted range [-pi/2, pi/2] (guaranteed by clip*pi*0.5).
// Branch-free polynomials, error ~6e-8 (sin, relative) / ~6e-9 (cos, absolute):
// no Payne-Hanek reduction, no divergence — unlike libm sinf/cosf.
__device__ __forceinline__ void sincos_hpi(float r, float& s, float& c) {
  const float r2 = r * r;
  float sp = -2.50521084e-08f;
  sp = fmaf(sp, r2,  2.75573192e-06f);
  sp = fmaf(sp, r2, -1.98412698e-04f);
  sp = fmaf(sp, r2,  8.33333333e-03f);
  sp = fmaf(sp, r2, -1.66666667e-01f);
  s  = fmaf(sp * r2, r, r);                 // r + r^3*(...)
  float cp =  2.08767570e-09f;
  cp = fmaf(cp, r2, -2.75573192e-07f);
  cp = fmaf(cp, r2,  2.48015873e-05f);
  cp = fmaf(cp, r2, -1.38888889e-03f);
  cp = fmaf(cp, r2,  4.16666667e-02f);
  cp = fmaf(cp, r2, -5.00000000e-01f);
  c  = fmaf(cp, r2, 1.0f);
}

// Simulate one 2-qubit pair (A = control wire, B = target wire), real amplitudes.
// State v[q], q = 2*iA + iB. Encoding RY layer folded into the product init.
// Layer = RY(A) ⊗ RY(B) then CNOT(A->B) (swap of the iA=1 sub-pair). 3 layers.
__device__ __forceinline__ void pair_sim(float cA, float sA, float cB, float sB,
                                         float p[4]) {
  float v0 = cA * cB, v1 = cA * sB, v2 = sA * cB, v3 = sA * sB;
#pragma unroll
  for (int l = 0; l < 3; ++l) {
    // RY on wire A: mixes (v0,v2) and (v1,v3)
    float t0 = cA * v0 - sA * v2;
    float t2 = sA * v0 + cA * v2;
    float t1 = cA * v1 - sA * v3;
    float t3 = sA * v1 + cA * v3;
    // RY on wire B: mixes (t0,t1) and (t2,t3)
    float u0 = cB * t0 - sB * t1;
    float u1 = sB * t0 + cB * t1;
    float u2 = cB * t2 - sB * t3;
    float u3 = sB * t2 + cB * t3;
    // CNOT(A->B): flip target where control==1  => swap v2 <-> v3
    v0 = u0; v1 = u1; v2 = u3; v3 = u2;
  }
  p[0] = v0 * v0; p[1] = v1 * v1; p[2] = v2 * v2; p[3] = v3 * v3;
}

__global__ __launch_bounds__(kBlock) void qconv_wmma_kernel(
    const float* __restrict__ patch, float* __restrict__ out, int n_elem) {
  // Per-wave staging: 32 elements x 8 probs, and 32 elements x 4 feats.
  __shared__ __align__(16) float ldsP[kWaves * kWave * 8];
  __shared__ __align__(16) float ldsO[kWaves * kWave * 4];

  const int lane = threadIdx.x & (kWave - 1);
  const int wave = threadIdx.x >> 5;
  const int e_base = blockIdx.x * kBlock + wave * kWave;
  int e = e_base + lane;
  if (e >= n_elem) e = n_elem - 1;  // keep full waves: WMMA requires EXEC==all-1s

  // ---- per-element circuit simulation (pure fp32 VALU, branch-free) ----
  const float4 x = reinterpret_cast<const float4*>(patch)[e];
  const float h0 = fminf(fmaxf(x.x, -1.f), 1.f) * kHalfPi;
  const float h1 = fminf(fmaxf(x.y, -1.f), 1.f) * kHalfPi;
  const float h2 = fminf(fmaxf(x.z, -1.f), 1.f) * kHalfPi;
  const float h3 = fminf(fmaxf(x.w, -1.f), 1.f) * kHalfPi;
  float c0, s0, c1, s1, c2, s2, c3, s3;
  sincos_hpi(h0, s0, c0);
  sincos_hpi(h1, s1, c1);
  sincos_hpi(h2, s2, c2);
  sincos_hpi(h3, s3, c3);

  float p01[4], p23[4];
  pair_sim(c0, s0, c1, s1, p01);  // wires 0,1
  pair_sim(c2, s2, c3, s3, p23);  // wires 2,3

  float* myP = &ldsP[(wave * kWave + lane) * 8];
#pragma unroll
  for (int k = 0; k < 4; ++k) { myP[k] = p01[k]; myP[4 + k] = p23[k]; }
  __syncthreads();

  // ---- WMMA fp32 16x16x4: feats(m,n) = sum_k probs(m,k) * Z(k,n) ----
  // A layout (16x4 fp32): lane L -> M = L%16; VGPR0 holds K = (L<16 ? 0 : 2),
  // VGPR1 holds K = (L<16 ? 1 : 3). B (4x16) mirrors this on the K axis.
  const int m = lane & 15;
  const int hi = lane >> 4;       // selects K = {0,1} vs {2,3}
  const int n = m;                // output column this lane carries in B/D
  // Z-sign matrix columns: col0/2 = (+,+,-,-) over q, col1/3 = (+,-,+,-).
  // With k0 = 2*hi, k1 = 2*hi+1:  zA[k0]=zA[k1]=(hi?-1:+1); zB[k0]=+1, zB[k1]=-1.
  const float zs = hi ? -1.f : 1.f;
  const v2f b01 = { (n == 0) ? zs : (n == 1) ?  1.f : 0.f,
                    (n == 0) ? zs : (n == 1) ? -1.f : 0.f };
  const v2f b23 = { (n == 2) ? zs : (n == 3) ?  1.f : 0.f,
                    (n == 2) ? zs : (n == 3) ? -1.f : 0.f };

#pragma unroll
  for (int t = 0; t < 2; ++t) {   // two 16-element M-tiles per wave
    const float* src = &ldsP[(wave * kWave + t * 16 + m) * 8];
    v2f a01 = { src[hi * 2 + 0],     src[hi * 2 + 1] };
    v2f a23 = { src[4 + hi * 2 + 0], src[4 + hi * 2 + 1] };
    v8f acc = {};
    acc = __builtin_amdgcn_wmma_f32_16x16x4_f32(false, a01, false, b01,
                                                (short)0, acc, false, false);
    acc = __builtin_amdgcn_wmma_f32_16x16x4_f32(false, a23, false, b23,
                                                (short)0, acc, false, false);
    // D layout: lane L, VGPR i -> row = i + 8*(L>=16), col = L%16.
    if (n < 4) {
#pragma unroll
      for (int i = 0; i < 8; ++i) {
        const int row = i + 8 * hi;
        ldsO[(wave * kWave + t * 16 + row) * 4 + n] = acc[i];
      }
    }
  }
  __syncthreads();

  // Coalesced float4 store: lane L owns element e_base+L's 4 features.
  const float4 r =
      reinterpret_cast<const float4*>(&ldsO[wave * kWave * 4])[lane];
  reinterpret_cast<float4*>(out)[e] = r;
}

extern "C" void kernel_launch(void* const* d_in, const int* in_sizes, int n_in,
                              void* d_out, int out_size, void* d_ws, size_t ws_size,
                              hipStream_t stream) {
  const float* patch = (const float*)d_in[0];   // (B, 2, 2) fp32, flat B*4
  float* out = (float*)d_out;                   // (B, 4) fp32
  const int n_elem = in_sizes[0] / 4;
  const int blocks = (n_elem + kBlock - 1) / kBlock;
  qconv_wmma_kernel<<<blocks, kBlock, 0, stream>>>(patch, out, n_elem);
}